// Shift_Block_split_45758581572309
// MI455X (gfx1250) — compile-verified
//
#include <hip/hip_runtime.h>
#include <math.h>

// Problem constants (from reference): x (16,3,1024,1024) f32, DIA=2.
#define BB  16
#define CC  3
#define H   1024
#define W   1024
#define OH  512
#define OW  512
#define TH  32            // output tile height (x_o pixels)
#define TW  32            // output tile width
// pooled x_o tile with dilation-2 halo: 36 rows, stride 40 (16B-aligned f4 rows)
#define XO_ROWS   36
#define XO_STRIDE 40
// staged x region: rows [2*ty0-5 .. 2*ty0+67] (73), cols [2*tx0-8 .. 2*tx0+71]
// (80, 16B-aligned base so every lane's 16B chunk is fully in- or out-of-image)
#define XS_ROWS   73
#define XS_STRIDE 80

#if defined(__has_builtin)
#if __has_builtin(__builtin_amdgcn_global_load_async_to_lds_b128) && \
    __has_builtin(__builtin_amdgcn_s_wait_asynccnt)
#define USE_ASYNC128 1
#endif
#endif
#ifndef USE_ASYNC128
#define USE_ASYNC128 0
#endif

typedef int v4i __attribute__((ext_vector_type(4)));
typedef __attribute__((address_space(1))) v4i gas_v4i;   // global
typedef __attribute__((address_space(3))) v4i las_v4i;   // LDS

__global__ __launch_bounds__(256)
void shift_block_fused_kernel(const float* __restrict__ x,
                              const float* __restrict__ w1,
                              const float* __restrict__ b1,
                              const float* __restrict__ w2,
                              const float* __restrict__ b2,
                              float* __restrict__ out)
{
    __shared__ alignas(16) float xs[XS_ROWS * XS_STRIDE];   // raw x tile (23.4 KB)
    __shared__ alignas(16) float xo[XO_ROWS * XO_STRIDE];   // pooled tile  (5.8 KB)

    const int tid = threadIdx.x;
    const int tx0 = blockIdx.x * TW;            // x_o tile origin
    const int ty0 = blockIdx.y * TH;
    const int bc  = blockIdx.z;                 // b*3 + c
    const int c   = bc % CC;

    const float w1c = w1[c], b1c = b1[c], w2c = w2[c], b2c = b2[c];

    const float* xch = x + (size_t)bc * (H * W);
    const int xr0  = 2 * ty0 - 5;               // first staged x row
    const int xc0a = 2 * tx0 - 8;               // first staged x col (16B aligned)

    // ------------- Stage A: global x -> LDS, async b128 per lane ----------
    {
        const int r0 = tid / 20;                // row within a 12-row pass
        const int cl = tid - r0 * 20;           // 16B column lane (0..19)
        if (r0 < 12) {
            const int  xca  = xc0a + 4 * cl;    // x col of this lane's 16B chunk
            const bool colv = (unsigned)xca < (unsigned)W;  // all-or-nothing
            float* lrow0 = &xs[4 * cl];
            for (int r = r0; r < XS_ROWS; r += 12) {
                const int  xr    = xr0 + r;
                const bool valid = colv && ((unsigned)xr < (unsigned)H);
                float* ldst = lrow0 + r * XS_STRIDE;
                if (valid) {
                    const float* g = xch + ((long)xr * W + xca);
#if USE_ASYNC128
                    __builtin_amdgcn_global_load_async_to_lds_b128(
                        (gas_v4i*)(void*)g, (las_v4i*)ldst, /*offset=*/0, /*cpol=*/0);
#else
                    *(float4*)ldst = *(const float4*)g;
#endif
                } else {
                    *(float4*)ldst = make_float4(0.f, 0.f, 0.f, 0.f);  // zero pad
                }
            }
        }
    }
#if USE_ASYNC128
    __builtin_amdgcn_s_wait_asynccnt(0);
#endif
    __syncthreads();

    // ------------- Stage B: avgpool(3,s2,p1) of y=x*w1+b1, relu ----------
    // reduce_window zero-pads y, so pooled = (w1*sum_x + b1*nvalid)/9.
    // 4 consecutive x_o outputs per thread; 3x9 tap window via 9 b128 loads.
    const float inv9 = 1.0f / 9.0f;
    for (int g = tid; g < 9 * XO_ROWS; g += 256) {          // 324 groups
        const int hy  = g / 9;
        const int hx4 = (g - hy * 9) * 4;                   // first of 4 outputs
        const int oy  = ty0 - 2 + hy;
        // taps for outputs hx4..hx4+3 live at xs cols 2*hx4+3 .. 2*hx4+11,
        // rows 2*hy .. 2*hy+2; base (even, mult of 4) = 2*hx4.
        const int base = (2 * hy) * XS_STRIDE + 2 * hx4;
        const float4 A0 = *(const float4*)&xs[base];
        const float4 A1 = *(const float4*)&xs[base + 4];
        const float4 A2 = *(const float4*)&xs[base + 8];
        const float4 B0 = *(const float4*)&xs[base + XS_STRIDE];
        const float4 B1 = *(const float4*)&xs[base + XS_STRIDE + 4];
        const float4 B2 = *(const float4*)&xs[base + XS_STRIDE + 8];
        const float4 C0 = *(const float4*)&xs[base + 2 * XS_STRIDE];
        const float4 C1 = *(const float4*)&xs[base + 2 * XS_STRIDE + 4];
        const float4 C2 = *(const float4*)&xs[base + 2 * XS_STRIDE + 8];
        // column sums over the 3 rows, cols c0..c0+8 (c0 = base+3)
        const float cs0 = A0.w + B0.w + C0.w;
        const float cs1 = A1.x + B1.x + C1.x;
        const float cs2 = A1.y + B1.y + C1.y;
        const float cs3 = A1.z + B1.z + C1.z;
        const float cs4 = A1.w + B1.w + C1.w;
        const float cs5 = A2.x + B2.x + C2.x;
        const float cs6 = A2.y + B2.y + C2.y;
        const float cs7 = A2.z + B2.z + C2.z;
        const float cs8 = A2.w + B2.w + C2.w;
        const float s[4] = { cs0 + cs1 + cs2, cs2 + cs3 + cs4,
                             cs4 + cs5 + cs6, cs6 + cs7 + cs8 };
        const bool  rowin = (unsigned)oy < (unsigned)OH;
        const float rv    = (oy == 0) ? 2.0f : 3.0f;
        float4 v;
        float* vp = (float*)&v;
        #pragma unroll
        for (int k = 0; k < 4; ++k) {
            const int ox = tx0 - 2 + hx4 + k;
            float r = 0.0f;                       // conv zero-pad outside x_o
            if (rowin && (unsigned)ox < (unsigned)OW) {
                const float cv = (ox == 0) ? 2.0f : 3.0f;
                r = fmaxf((w1c * s[k] + b1c * (rv * cv)) * inv9, 0.0f);
            }
            vp[k] = r;
        }
        *(float4*)&xo[hy * XO_STRIDE + hx4] = v;
    }
    __syncthreads();

    // ------------- Stage C: dilated(2) 3x3 box-sum, gate, store ----------
    // One thread = 4 consecutive outputs; 6 b128 LDS loads; b128 store.
    {
        const int ly  = tid >> 3;                 // 0..31
        const int lx4 = (tid & 7) << 2;           // 0,4,...,28
        const int b0  = ly * XO_STRIDE + lx4;
        const float4 T0a = *(const float4*)&xo[b0];
        const float4 T0b = *(const float4*)&xo[b0 + 4];
        const float4 T1a = *(const float4*)&xo[b0 + 2 * XO_STRIDE];
        const float4 T1b = *(const float4*)&xo[b0 + 2 * XO_STRIDE + 4];
        const float4 T2a = *(const float4*)&xo[b0 + 4 * XO_STRIDE];
        const float4 T2b = *(const float4*)&xo[b0 + 4 * XO_STRIDE + 4];
        const float d0 = T0a.x + T1a.x + T2a.x;
        const float d1 = T0a.y + T1a.y + T2a.y;
        const float d2 = T0a.z + T1a.z + T2a.z;
        const float d3 = T0a.w + T1a.w + T2a.w;
        const float d4 = T0b.x + T1b.x + T2b.x;
        const float d5 = T0b.y + T1b.y + T2b.y;
        const float d6 = T0b.z + T1b.z + T2b.z;
        const float d7 = T0b.w + T1b.w + T2b.w;
        const float s9[4]  = { d0 + d2 + d4, d1 + d3 + d5,
                               d2 + d4 + d6, d3 + d5 + d7 };
        const float ctr[4] = { T1a.z, T1a.w, T1b.x, T1b.y };  // centers (col +2)
        float4 o;
        float* op = (float*)&o;
        #pragma unroll
        for (int k = 0; k < 4; ++k) {
            const float t = s9[k] * w2c + b2c;
            op[k] = ctr[k] / (1.0f + __expf(-t)) * 0.0f + ctr[k] * (1.0f / (1.0f + __expf(-t)));
        }
        // (the first term above is dead; keep the simple form:)
        #pragma unroll
        for (int k = 0; k < 4; ++k) {
            const float t = s9[k] * w2c + b2c;
            op[k] = ctr[k] * (1.0f / (1.0f + __expf(-t)));
        }
        float* och = out + (size_t)bc * (OH * OW);
        *(float4*)&och[(size_t)(ty0 + ly) * OW + (tx0 + lx4)] = o;
    }
}

extern "C" void kernel_launch(void* const* d_in, const int* in_sizes, int n_in,
                              void* d_out, int out_size, void* d_ws, size_t ws_size,
                              hipStream_t stream) {
    const float* x  = (const float*)d_in[0];
    const float* w1 = (const float*)d_in[1];
    const float* b1 = (const float*)d_in[2];
    const float* w2 = (const float*)d_in[3];
    const float* b2 = (const float*)d_in[4];
    float* out = (float*)d_out;

    dim3 grid(OW / TW, OH / TH, BB * CC);   // 16 x 16 x 48
    shift_block_fused_kernel<<<grid, 256, 0, stream>>>(x, w1, b1, w2, b2, out);
}